// RelativePositionEncoding_50010599195179
// MI455X (gfx1250) — compile-verified
//
#include <hip/hip_runtime.h>

#define LSEQ 1024
#define DIM  128
#define RMAX 32
#define SMAX 2
#define CR   (2 * RMAX + 2)        // 66
#define CS   (2 * SMAX + 2)        // 6
#define WROWS (2 * CR + 1 + CS)    // 139 rows: [0,66) si | [66,132) ti | 132 ent | [133,139) sym

typedef __attribute__((ext_vector_type(4))) float f4;
typedef __attribute__((ext_vector_type(4))) int   i4;

static __device__ __forceinline__ int iclip(int v, int lo, int hi) {
    return v < lo ? lo : (v > hi ? hi : v);
}

__global__ __launch_bounds__(256) void rpe_kernel(
    const int* __restrict__ seq_index, const int* __restrict__ seq_color,
    const int* __restrict__ seq_sym,   const int* __restrict__ seq_entity,
    const int* __restrict__ token_index, const float* __restrict__ W,
    float* __restrict__ out)
{
    // 139*128*4 = 71,168 B W table + 4 KB packed bins -> ~75 KB LDS, 4 WGs/WGP
    __shared__ float Wlds[WROWS * DIM];
    __shared__ int   bins[LSEQ];

    // ---- Issue async global->LDS DMA for W (ASYNCcnt path), don't wait yet ----
#if __has_builtin(__builtin_amdgcn_global_load_async_to_lds_b128)
    {
        const int n16 = (WROWS * DIM) / 4;   // 4448 16-byte chunks
        __attribute__((address_space(1))) char* gsrc =
            (__attribute__((address_space(1))) char*)(float*)W;
        __attribute__((address_space(3))) char* ldst =
            (__attribute__((address_space(3))) char*)Wlds;
        for (int t = threadIdx.x; t < n16; t += 256) {
            __builtin_amdgcn_global_load_async_to_lds_b128(
                (__attribute__((address_space(1))) i4*)(gsrc + 16 * t),
                (__attribute__((address_space(3))) i4*)(ldst + 16 * t),
                /*offset=*/0, /*cpol=*/0);
        }
    }
#else
    for (int t = threadIdx.x; t < (WROWS * DIM) / 4; t += 256)
        ((f4*)Wlds)[t] = ((const f4*)W)[t];
#endif

    const int i = blockIdx.x;            // one row i per block
    const int si_i = seq_index[i];
    const int sc_i = seq_color[i];
    const int ss_i = seq_sym[i];
    const int se_i = seq_entity[i];
    const int ti_i = token_index[i];

    // ---- Precompute packed W-row descriptors for all j (overlaps the DMA) ----
    // byte0: si row (0..65)  byte1: ti row (66..131)
    // byte2: sym row (133..138)  byte3: entity flag (0/1)
    for (int j = threadIdx.x; j < LSEQ; j += 256) {
        const int sj = seq_index[j];
        const int cj = seq_color[j];
        const int yj = seq_sym[j];
        const int ej = seq_entity[j];
        const int tj = token_index[j];

        const bool bc = (sc_i == cj);
        const bool bi = (si_i == sj);
        const bool be = (se_i == ej);

        const int r_si = bc ? (iclip(si_i - sj, -RMAX, RMAX) + RMAX) : (2 * RMAX + 1);
        const int r_ti = CR + ((bc && bi) ? (iclip(ti_i - tj, -RMAX, RMAX) + RMAX)
                                          : (2 * RMAX + 1));
        const int r_sy = (2 * CR + 1) + (be ? (iclip(ss_i - yj, -SMAX, SMAX) + SMAX)
                                            : (2 * SMAX + 1));
        bins[j] = r_si | (r_ti << 8) | (r_sy << 16) | ((be ? 1 : 0) << 24);
    }

#if __has_builtin(__builtin_amdgcn_global_load_async_to_lds_b128)
    asm volatile("s_wait_asynccnt 0x0" ::: "memory");
#endif
    __syncthreads();

    const int wave = threadIdx.x >> 5;   // 8 waves per block
    const int lane = threadIdx.x & 31;   // wave32
    const int coff = lane * 4;           // each lane: 4 consecutive dims (float4)

    // entity row (132) is fixed: keep its slice in registers for the whole kernel
    const f4 ent = *(const f4*)&Wlds[(2 * CR) * DIM + coff];

    for (int j = wave; j < LSEQ; j += 8) {
        const int pk  = bins[j];                 // broadcast ds_load_b32
        const int rsi =  pk        & 0xFF;
        const int rti = (pk >> 8)  & 0xFF;
        const int rsy = (pk >> 16) & 0xFF;
        const float em = (float)(pk >> 24);      // 0.0 or 1.0

        // wave-uniform row, lane-strided columns -> conflict-free ds_load_b128
        const f4 a = *(const f4*)&Wlds[rsi * DIM + coff];
        const f4 b = *(const f4*)&Wlds[rti * DIM + coff];
        const f4 s = *(const f4*)&Wlds[rsy * DIM + coff];

        f4 r;
        r.x = (a.x + b.x) + (s.x + em * ent.x);
        r.y = (a.y + b.y) + (s.y + em * ent.y);
        r.z = (a.z + b.z) + (s.z + em * ent.z);
        r.w = (a.w + b.w) + (s.w + em * ent.w);

        // streaming 512 MB output: non-temporal 128-bit stores, fully coalesced
        f4* dst = (f4*)(out + ((size_t)i * LSEQ + (size_t)j) * DIM) + lane;
        __builtin_nontemporal_store(r, dst);
    }
}

extern "C" void kernel_launch(void* const* d_in, const int* in_sizes, int n_in,
                              void* d_out, int out_size, void* d_ws, size_t ws_size,
                              hipStream_t stream) {
    const int*   seq_index   = (const int*)d_in[0];
    const int*   seq_color   = (const int*)d_in[1];
    const int*   seq_sym     = (const int*)d_in[2];
    const int*   seq_entity  = (const int*)d_in[3];
    const int*   token_index = (const int*)d_in[4];
    const float* W           = (const float*)d_in[5];
    float*       out         = (float*)d_out;

    rpe_kernel<<<dim3(LSEQ), dim3(256), 0, stream>>>(
        seq_index, seq_color, seq_sym, seq_entity, token_index, W, out);
}